// DisenGCN_2259152797853
// MI455X (gfx1250) — compile-verified
//
#include <hip/hip_runtime.h>
#include <math.h>

typedef __attribute__((ext_vector_type(2))) float v2f;
typedef __attribute__((ext_vector_type(8))) float v8f;

#define N_NODES 20000
#define MNB     16
#define FEAT    500
#define OUTD    656
#define ROUTIT  5

// -------- 16-lane (half-wave) reduction via lane shuffles (wave32) --------
__device__ __forceinline__ float hsum16(float v) {
    v += __shfl_xor(v, 1, 16);
    v += __shfl_xor(v, 2, 16);
    v += __shfl_xor(v, 4, 16);
    v += __shfl_xor(v, 8, 16);
    return v;
}

// ---------------------------------------------------------------------------
// GEMM: C[m][n] = relu?(sum_k A[m][k] * W[n][k] + bias[n])
// W is [fout x fin] row-major (i.e. we compute A @ W^T + b), FIN compile-time
// so every B-tile load is a global_load_b64 with an immediate offset off one
// address VGPR pair (no v_add_nc_u64 chains, loads clause together and the
// WMMA burst overlaps the next loads via descending s_wait_loadcnt).
// One wave computes a 16-row stripe across all NT*16 output columns using
// V_WMMA_F32_16X16X4_F32.  Fragment layouts per CDNA5 ISA 7.12.2:
//   A 16x4 fp32 : lanes 0-15 -> M=lane, VGPR{0,1}=K{0,1}; lanes 16-31 K{2,3}
//   B 4x16 fp32 : lanes 0-15 -> N=lane, VGPR{0,1}=K{0,1}; lanes 16-31 K{2,3}
//   C/D 16x16   : VGPR r -> M=r (lanes 0-15) / M=r+8 (lanes 16-31), N=lane%16
// ---------------------------------------------------------------------------
template<int NT, int FIN, bool RELU>
__global__ __launch_bounds__(128) void gemm_wmma_kernel(
    const float* __restrict__ A, int lda,
    const float* __restrict__ W,
    const float* __restrict__ bias,
    float* __restrict__ C, int ldc, int nrows)
{
    const int lane = threadIdx.x & 31;
    const int wv   = threadIdx.x >> 5;
    const int m0   = (blockIdx.x * 4 + wv) * 16;
    if (m0 >= nrows) return;               // wave-uniform: EXEC stays all-ones

    const int r16 = lane & 15;
    const int hf  = lane >> 4;

    const float* aptr = A + (size_t)(m0 + r16) * lda + 2 * hf;
    const float* wrow = W + r16 * FIN + 2 * hf;      // tile j at +j*16*FIN

    const v8f zero = {0.f, 0.f, 0.f, 0.f, 0.f, 0.f, 0.f, 0.f};
    v8f acc[NT];
#pragma unroll
    for (int j = 0; j < NT; ++j) acc[j] = zero;

#pragma unroll 4
    for (int kb = 0; kb < FIN; kb += 4) {
        __builtin_prefetch(aptr + 32, 0, 3);         // near-cache prefetch
        // issue all fragment loads first, then the WMMA burst
        v2f a = *(const v2f*)aptr;
        v2f b[NT];
#pragma unroll
        for (int j = 0; j < NT; ++j)
            b[j] = *(const v2f*)(wrow + j * 16 * FIN);   // immediate offsets
        aptr += 4;
        wrow += 4;
#pragma unroll
        for (int j = 0; j < NT; ++j) {
            // D = A*B + C   (fp32 16x16x4 WMMA)
            acc[j] = __builtin_amdgcn_wmma_f32_16x16x4_f32(
                /*neg_a=*/false, a, /*neg_b=*/false, b[j],
                /*c_mod=*/(short)0, acc[j],
                /*reuse_a=*/false, /*reuse_b=*/false);
        }
    }

#pragma unroll
    for (int j = 0; j < NT; ++j) {
        const int c  = j * 16 + r16;
        const float bv = bias[c];
        float* crow = C + (size_t)(m0 + hf * 8) * ldc + c;
#pragma unroll
        for (int r = 0; r < 8; ++r) {
            float v = acc[j][r] + bv;
            if (RELU) v = fmaxf(v, 0.0f);
            crow[(size_t)r * ldc] = v;
        }
    }
}

// ---------------------------------------------------------------------------
// Per-capsule L2 normalize: out[n][k][d] = in / max(||in[n][k][:]||, 1e-12)
// 16-lane groups align exactly with d (k16 is a multiple of 16).
// ---------------------------------------------------------------------------
__global__ __launch_bounds__(256) void l2norm_kernel(
    const float* __restrict__ in, int ldin,
    float* __restrict__ out, int k16, int total)
{
    int t = blockIdx.x * 256 + threadIdx.x;
    if (t >= total) return;
    int nrow = t / k16;
    int f    = t - nrow * k16;
    float v  = in[(size_t)nrow * ldin + f];
    float s  = hsum16(v * v);
    out[t]   = v * (1.0f / fmaxf(sqrtf(s), 1e-12f));
}

// ---------------------------------------------------------------------------
// Disentangled routing, one wave32 per node.
//   lanes: d = lane%16, half = lane/16; the two halves process neighbor pairs
//   z (16 neighbors x K x 16) staged in LDS; u,x replicated across halves.
// 5 iterations: p = <z,u>_d ; softmax_k ; u = sum_m z*p + x ; l2norm (not last)
// Final: relu(u) -> strided slice of d_out.
// ---------------------------------------------------------------------------
template<int K>
__global__ __launch_bounds__(128) void routing_kernel(
    const float* __restrict__ xnorm,      // [N][K*16] contiguous, normalized
    const int*   __restrict__ nb,         // [N][16]
    float* __restrict__ out, int ldout)   // pre-offset to column slice
{
    constexpr int K16 = K * 16;
    constexpr int C4  = K16 / 4;          // float4 chunks per neighbor row
    __shared__ __align__(16) float zsh[4][MNB * K16];

    const int lane = threadIdx.x & 31;
    const int wv   = threadIdx.x >> 5;
    const int node = blockIdx.x * 4 + wv; // grid is exact: N_NODES/4 blocks
    const int d    = lane & 15;
    const int hf   = lane >> 4;

    float* z = zsh[wv];
    const int* nbp = nb + node * MNB;

    // gather 16 neighbors' normalized capsule vectors into LDS (b128 copies)
    {
        const float4* xn4 = (const float4*)xnorm;
        float4* z4 = (float4*)z;
        for (int idx = lane; idx < MNB * C4; idx += 32) {
            int m  = idx / C4;
            int f4 = idx - m * C4;
            z4[m * C4 + f4] = xn4[(size_t)nbp[m] * C4 + f4];
        }
    }

    float x[K], u[K];
#pragma unroll
    for (int k = 0; k < K; ++k) {
        x[k] = xnorm[(size_t)node * K16 + k * 16 + d];
        u[k] = x[k];
    }
    __syncthreads();

    for (int it = 0; it < ROUTIT; ++it) {
        float unew[K];
#pragma unroll
        for (int k = 0; k < K; ++k) unew[k] = 0.0f;

        for (int m2 = 0; m2 < MNB / 2; ++m2) {
            const float* zm = z + (m2 * 2 + hf) * K16;   // this half's neighbor
            float zv[K], p[K];
#pragma unroll
            for (int k = 0; k < K; ++k) {
                zv[k] = zm[k * 16 + d];
                p[k]  = hsum16(zv[k] * u[k]);            // agreement <z,u>_d
            }
            float mx = p[0];
#pragma unroll
            for (int k = 1; k < K; ++k) mx = fmaxf(mx, p[k]);
            float s = 0.0f;
#pragma unroll
            for (int k = 0; k < K; ++k) { p[k] = __expf(p[k] - mx); s += p[k]; }
            float inv = 1.0f / s;                        // softmax over k
#pragma unroll
            for (int k = 0; k < K; ++k) unew[k] += zv[k] * (p[k] * inv);
        }

#pragma unroll
        for (int k = 0; k < K; ++k) {
            float t2 = unew[k] + __shfl_xor(unew[k], 16, 32);  // combine halves
            t2 += x[k];                                        // + self
            if (it < ROUTIT - 1) {
                float s = hsum16(t2 * t2);
                u[k] = t2 * (1.0f / fmaxf(sqrtf(s), 1e-12f));
            } else {
                u[k] = t2;
            }
        }
    }

    if (hf == 0) {
#pragma unroll
        for (int k = 0; k < K; ++k)
            out[(size_t)node * ldout + k * 16 + d] = fmaxf(u[k], 0.0f);
    }
}

// ---------------------------------------------------------------------------
extern "C" void kernel_launch(void* const* d_in, const int* in_sizes, int n_in,
                              void* d_out, int out_size, void* d_ws, size_t ws_size,
                              hipStream_t stream)
{
    const float* feature = (const float*)d_in[0];
    const int*   nb      = (const int*)d_in[1];
    const float* pca_w   = (const float*)d_in[2];
    const float* pca_b   = (const float*)d_in[3];
    const float* w[6] = {nullptr, (const float*)d_in[4], (const float*)d_in[6],
                         (const float*)d_in[8], (const float*)d_in[10], (const float*)d_in[12]};
    const float* b[6] = {nullptr, (const float*)d_in[5], (const float*)d_in[7],
                         (const float*)d_in[9], (const float*)d_in[11], (const float*)d_in[13]};
    float* out   = (float*)d_out;
    float* xbuf  = (float*)d_ws;                       // [N][<=128] linear out
    float* xnorm = xbuf + (size_t)N_NODES * 128;       // [N][<=128] normalized

    const int caps[6]   = {8, 7, 6, 5, 4, 3};
    const int colofs[7] = {0, 128, 256, 368, 464, 544, 608};

    const int gemmBlocks  = (N_NODES / 16 + 3) / 4;    // 313 (4 waves/block)
    const int routeBlocks = N_NODES / 4;               // 5000 (exact)

    // SparseInput: xs[0] = relu(feature @ pca_w^T + pca_b) -> d_out cols 0..127
    gemm_wmma_kernel<8, FEAT, true><<<gemmBlocks, 128, 0, stream>>>(
        feature, FEAT, pca_w, pca_b, out + colofs[0], OUTD, N_NODES);

    for (int i = 0; i < 6; ++i) {
        const int K   = caps[i];
        const int K16 = K * 16;
        const float* nin;
        int ldin;
        if (i == 0) {
            nin = out + colofs[0];
            ldin = OUTD;
        } else {
            // NT == K since fout = K*16; FIN = caps[i-1]*16 (compile-time)
            switch (K) {
            case 7: gemm_wmma_kernel<7, 128, false><<<gemmBlocks, 128, 0, stream>>>(
                        out + colofs[i], OUTD, w[i], b[i], xbuf, K16, N_NODES); break;
            case 6: gemm_wmma_kernel<6, 112, false><<<gemmBlocks, 128, 0, stream>>>(
                        out + colofs[i], OUTD, w[i], b[i], xbuf, K16, N_NODES); break;
            case 5: gemm_wmma_kernel<5,  96, false><<<gemmBlocks, 128, 0, stream>>>(
                        out + colofs[i], OUTD, w[i], b[i], xbuf, K16, N_NODES); break;
            case 4: gemm_wmma_kernel<4,  80, false><<<gemmBlocks, 128, 0, stream>>>(
                        out + colofs[i], OUTD, w[i], b[i], xbuf, K16, N_NODES); break;
            case 3: gemm_wmma_kernel<3,  64, false><<<gemmBlocks, 128, 0, stream>>>(
                        out + colofs[i], OUTD, w[i], b[i], xbuf, K16, N_NODES); break;
            }
            nin = xbuf;
            ldin = K16;
        }

        const int total = N_NODES * K16;
        l2norm_kernel<<<(total + 255) / 256, 256, 0, stream>>>(nin, ldin, xnorm, K16, total);

        switch (K) {
        case 8: routing_kernel<8><<<routeBlocks, 128, 0, stream>>>(xnorm, nb, out + colofs[i + 1], OUTD); break;
        case 7: routing_kernel<7><<<routeBlocks, 128, 0, stream>>>(xnorm, nb, out + colofs[i + 1], OUTD); break;
        case 6: routing_kernel<6><<<routeBlocks, 128, 0, stream>>>(xnorm, nb, out + colofs[i + 1], OUTD); break;
        case 5: routing_kernel<5><<<routeBlocks, 128, 0, stream>>>(xnorm, nb, out + colofs[i + 1], OUTD); break;
        case 4: routing_kernel<4><<<routeBlocks, 128, 0, stream>>>(xnorm, nb, out + colofs[i + 1], OUTD); break;
        case 3: routing_kernel<3><<<routeBlocks, 128, 0, stream>>>(xnorm, nb, out + colofs[i + 1], OUTD); break;
        }
    }
}